// GraphConvolutionLayer_4947802325636
// MI455X (gfx1250) — compile-verified
//
#include <hip/hip_runtime.h>
#include <math.h>

typedef float v2f __attribute__((ext_vector_type(2)));
typedef float v8f __attribute__((ext_vector_type(8)));

#define Bsz 4096
#define Nn 32
#define Ee 5
#define Ff 64
#define Uu 128

// ---- LDS layout (in floats), padded strides chosen for bank-conflict-free WMMA operand access ----
#define NODE_STRIDE 68                       // 32x64 node tile; (68*m + k) % 64 = (4m+k) -> conflict free
#define ADJ_STRIDE  36                       // 36*i mod 64 is a full 16-cycle -> conflict free
#define HID_STRIDE  140                      // 140%64=12: store rows +8 -> +32 banks; read rows +2 -> +24 banks
#define NODE_OFF 0
#define ADJ_OFF  (Nn * NODE_STRIDE)                    // 2176
#define ADJ_E    (Nn * ADJ_STRIDE)                     // 1152
#define HID_OFF  (ADJ_OFF + 4 * ADJ_E)                 // 6784
#define HID_E    (Nn * HID_STRIDE)                     // 4480
#define SMEM_FLOATS (HID_OFF + 5 * HID_E)              // 29184 floats = 116736 bytes

extern "C" __global__ __launch_bounds__(256)
void gcn_fused_kernel(const float* __restrict__ adjacency,  // [B,N,N,E]
                      const float* __restrict__ node,       // [B,N,F]
                      const float* __restrict__ W1,         // [4,F,U]
                      const float* __restrict__ b1,         // [4,U]
                      const float* __restrict__ W2,         // [F,U]
                      const float* __restrict__ b2,         // [U]
                      float* __restrict__ out_adj,          // mirror of adjacency
                      float* __restrict__ out_node,         // mirror of node
                      float* __restrict__ out)              // [B,N,U]
{
    extern __shared__ float smem[];
    const int b    = blockIdx.x;
    const int tid  = threadIdx.x;
    const int lane = tid & 31;
    const int wave = tid >> 5;
    const int half = lane >> 4;     // 0: lanes 0-15, 1: lanes 16-31
    const int l16  = lane & 15;

    // ================= Stage 0: global -> LDS, fused with tuple mirror copies =================
    {
        // node: 32*64 = 2048 floats = 512 float4; 2 per thread
        const float4* nsrc = (const float4*)(node + (size_t)b * Nn * Ff);
        float4*       ndst = (float4*)(out_node + (size_t)b * Nn * Ff);
        #pragma unroll
        for (int it = 0; it < 2; ++it) {
            int v = tid + it * 256;
            float4 d = nsrc[v];
            ndst[v] = d;
            int fidx = v * 4;
            int row  = fidx >> 6;          // / 64
            int col  = fidx & 63;
            *(float4*)&smem[NODE_OFF + row * NODE_STRIDE + col] = d;
        }
        // adjacency block: 32*32*5 = 5120 floats = 1280 float4; 5 per thread
        const float4* asrc = (const float4*)(adjacency + (size_t)b * Nn * Nn * Ee);
        float4*       adst = (float4*)(out_adj + (size_t)b * Nn * Nn * Ee);
        #pragma unroll
        for (int it = 0; it < 5; ++it) {
            int v = tid + it * 256;
            float4 d = asrc[v];
            adst[v] = d;
            int q0 = v * 4;
            float vals[4] = {d.x, d.y, d.z, d.w};
            #pragma unroll
            for (int c2 = 0; c2 < 4; ++c2) {
                int q = q0 + c2;
                int e = q % 5;             // edge-type index in memory layout
                if (e != 0) {              // drop edge type 0
                    int ij = q / 5;
                    int i  = ij >> 5;
                    int j  = ij & 31;
                    smem[ADJ_OFF + (e - 1) * ADJ_E + i * ADJ_STRIDE + j] = vals[c2];
                }
            }
        }
    }
    __syncthreads();

    // ====== Stage 1: hidden[e] = node @ W1[e] + b1[e]  (e=0..3);  e==4: self = node @ W2 + b2 ======
    // 40 strips = 5 weight-matrices x 8 N-tiles; each strip: 2 M-tiles x 16 K-steps of WMMA f32 16x16x4
    for (int s = wave; s < 40; s += 8) {
        int e  = s >> 3;                   // 0..4
        int nt = s & 7;
        int u  = nt * 16 + l16;
        const float* Wp  = (e < 4) ? (W1 + (size_t)e * Ff * Uu) : W2;
        float bias       = (e < 4) ? b1[e * Uu + u] : b2[u];
        #pragma unroll
        for (int mt = 0; mt < 2; ++mt) {
            v8f acc;
            #pragma unroll
            for (int r = 0; r < 8; ++r) acc[r] = bias;     // bias depends only on column u
            const float* arow = &smem[NODE_OFF + (mt * 16 + l16) * NODE_STRIDE];
            #pragma unroll
            for (int k0 = 0; k0 < Ff; k0 += 4) {
                int k = k0 + half * 2;
                float2 a2 = *(const float2*)&arow[k];                  // A: M=l16, K={k,k+1}
                v2f a  = {a2.x, a2.y};
                v2f bb = {Wp[(size_t)k * Uu + u],                      // B: K rows, N=u
                          Wp[(size_t)(k + 1) * Uu + u]};
                acc = __builtin_amdgcn_wmma_f32_16x16x4_f32(
                          false, a, false, bb, (short)0, acc, false, false);
            }
            float* hdst = &smem[HID_OFF + e * HID_E + (mt * 16 + half * 8) * HID_STRIDE + u];
            #pragma unroll
            for (int r = 0; r < 8; ++r) hdst[r * HID_STRIDE] = acc[r]; // C row = half*8 + r
        }
    }
    __syncthreads();

    // ====== Stage 2: out = tanh( sum_e adj_e @ hidden[e] + self ) ======
    // 16 output tiles (2 M x 8 N); init C from self term, accumulate 4 edge types x 8 K-steps
    for (int t = wave; t < 16; t += 8) {
        int mt = t >> 3;
        int nt = t & 7;
        int u  = nt * 16 + l16;
        v8f c;
        const float* selfp = &smem[HID_OFF + 4 * HID_E + (mt * 16 + half * 8) * HID_STRIDE + u];
        #pragma unroll
        for (int r = 0; r < 8; ++r) c[r] = selfp[r * HID_STRIDE];
        #pragma unroll
        for (int e = 0; e < 4; ++e) {
            const float* abase = &smem[ADJ_OFF + e * ADJ_E + (mt * 16 + l16) * ADJ_STRIDE];
            const float* hbase = &smem[HID_OFF + e * HID_E + u];
            #pragma unroll
            for (int k0 = 0; k0 < Nn; k0 += 4) {
                int k = k0 + half * 2;
                float2 a2 = *(const float2*)&abase[k];                 // A: adj[i][k..k+1]
                v2f a  = {a2.x, a2.y};
                v2f bb = {hbase[k * HID_STRIDE], hbase[(k + 1) * HID_STRIDE]};
                c = __builtin_amdgcn_wmma_f32_16x16x4_f32(
                        false, a, false, bb, (short)0, c, false, false);
            }
        }
        float* obase = out + (size_t)b * Nn * Uu + (mt * 16 + half * 8) * Uu + u;
        #pragma unroll
        for (int r = 0; r < 8; ++r) obase[r * Uu] = tanhf(c[r]);
    }
}

extern "C" void kernel_launch(void* const* d_in, const int* in_sizes, int n_in,
                              void* d_out, int out_size, void* d_ws, size_t ws_size,
                              hipStream_t stream) {
    (void)in_sizes; (void)n_in; (void)out_size; (void)d_ws; (void)ws_size;
    const float* adjacency = (const float*)d_in[0];
    const float* node      = (const float*)d_in[1];
    const float* W1        = (const float*)d_in[2];
    const float* b1        = (const float*)d_in[3];
    const float* W2        = (const float*)d_in[4];
    const float* b2        = (const float*)d_in[5];

    float* out_base = (float*)d_out;
    float* out_adj  = out_base;                                       // 4096*32*32*5
    float* out_node = out_base + (size_t)Bsz * Nn * Nn * Ee;          // 4096*32*64
    float* out_y    = out_node + (size_t)Bsz * Nn * Ff;               // 4096*32*128

    size_t smem_bytes = (size_t)SMEM_FLOATS * sizeof(float);
    hipLaunchKernelGGL(gcn_fused_kernel, dim3(Bsz), dim3(256), smem_bytes, stream,
                       adjacency, node, W1, b1, W2, b2, out_adj, out_node, out_y);
}